// DualBranchGNN_Deep_87024627352104
// MI455X (gfx1250) — compile-verified
//
#include <hip/hip_runtime.h>
#include <math.h>

typedef float v2f __attribute__((ext_vector_type(2)));
typedef float v8f __attribute__((ext_vector_type(8)));

#define HID 64
#define MT  4   // 16-row M sub-tiles per wave in the WMMA GEMM (64-row strip)

__device__ __forceinline__ float eluf(float x)     { return x > 0.f ? x : (expf(x) - 1.f); }
__device__ __forceinline__ float sigm(float x)     { return 1.f / (1.f + expf(-x)); }
__device__ __forceinline__ float lrelu02(float x)  { return x > 0.f ? x : 0.2f * x; }

// monotone float<->uint encoding for atomicMax on floats
__device__ __forceinline__ unsigned enc_key(float f) {
    unsigned b = __float_as_uint(f);
    return (b & 0x80000000u) ? ~b : (b | 0x80000000u);
}
__device__ __forceinline__ float dec_key(unsigned k) {
    unsigned b = (k & 0x80000000u) ? (k & 0x7fffffffu) : ~k;
    return __uint_as_float(b);
}

// ---------------------------------------------------------------- fill
__global__ void fill_kernel(float* __restrict__ p, float v, long n) {
    long i = (long)blockIdx.x * blockDim.x + threadIdx.x;
    if (i < n) p[i] = v;
}

// ---------------------------------------------------------------- edge weights
__global__ void edge_weight_kernel(const int* __restrict__ ei,
                                   const float* __restrict__ noise,
                                   const float* __restrict__ coord,
                                   const float* __restrict__ w1,   // (10,10) row-major
                                   const float* __restrict__ b1,   // (10)
                                   const float* __restrict__ w2,   // (10)
                                   const float* __restrict__ b2,   // (1)
                                   float* __restrict__ dyn, int E) {
    int e = blockIdx.x * blockDim.x + threadIdx.x;
    if (e >= E) return;
    int s = ei[e], d = ei[E + e];
    float nf[10];
#pragma unroll
    for (int k = 0; k < 10; ++k) nf[k] = noise[(long)s * 10 + k];
    float ni = b2[0];
#pragma unroll
    for (int j = 0; j < 10; ++j) {
        float acc = b1[j];
#pragma unroll
        for (int k = 0; k < 10; ++k) acc += nf[k] * w1[j * 10 + k]; // noise @ W1^T
        ni += eluf(acc) * w2[j];
    }
    float dx = coord[(long)s * 2 + 0] - coord[(long)d * 2 + 0];
    float dy = coord[(long)s * 2 + 1] - coord[(long)d * 2 + 1];
    float gw = expf(-(dx * dx + dy * dy) * (1.f / (2.f * 30.f * 30.f)));
    dyn[e] = 0.1f + 1.9f * sigm(gw * (1.f + ni) - 1.f);
}

// ---------------------------------------------------------------- WMMA GEMM
// Out[nRows x dout] = act( concat(A0,A1)[nRows x din] @ W[din x dout] + bias )
// Virtual concat: cols [0,splitK) from A0 (lda0), rest from A1 (lda1); splitK % 4 == 0.
// act: 0=none 1=relu 2=elu 3=sigmoid.
// One wave32 per 64x16 output strip: 4 independent accumulator chains share B loads.
__global__ void gemm_wmma_kernel(const float* __restrict__ A0, const float* __restrict__ A1,
                                 int splitK, int lda0, int lda1,
                                 const float* __restrict__ W, const float* __restrict__ bias,
                                 float* __restrict__ Out, int nRows, int din, int dout, int act) {
    int lane = threadIdx.x & 31;
    int wv   = threadIdx.x >> 5;
    int tilesN = dout >> 4;
    int tilesM = (nRows + 16 * MT - 1) / (16 * MT);
    int tile = blockIdx.x * (blockDim.x >> 5) + wv;
    if (tile >= tilesM * tilesN) return;            // wave-uniform exit: EXEC all-1 at WMMA
    int tm = tile / tilesN, tn = tile - tm * tilesN;
    int li = lane & 15, hi = lane >> 4;
    int nc = tn * 16 + li;

    int mrow[MT];                                    // A row per sub-tile (clamped reads)
#pragma unroll
    for (int t = 0; t < MT; ++t) {
        int m = tm * (16 * MT) + t * 16 + li;
        mrow[t] = m < nRows ? m : nRows - 1;
    }
    const v8f vzero = {0.f, 0.f, 0.f, 0.f, 0.f, 0.f, 0.f, 0.f};
    v8f acc[MT];
#pragma unroll
    for (int t = 0; t < MT; ++t) acc[t] = vzero;

#if __has_builtin(__builtin_amdgcn_wmma_f32_16x16x4_f32)
    int kEnd0 = splitK < din ? splitK : din;
    // phase 0: columns from A0
    for (int k0 = 0; k0 < kEnd0; k0 += 4) {
        int kb = k0 + hi * 2;   // lanes 0-15: K=k0,k0+1 ; lanes 16-31: K=k0+2,k0+3
        v2f b; b.x = W[(long)kb * dout + nc]; b.y = W[(long)(kb + 1) * dout + nc];
#pragma unroll
        for (int t = 0; t < MT; ++t) {
            const float* ap = A0 + (long)mrow[t] * lda0 + kb;
            v2f a; a.x = ap[0]; a.y = ap[1];
            acc[t] = __builtin_amdgcn_wmma_f32_16x16x4_f32(false, a, false, b, (short)0,
                                                           acc[t], false, false);
        }
    }
    // phase 1: columns from A1 (concat tail)
    for (int k0 = kEnd0; k0 < din; k0 += 4) {
        int kb = k0 + hi * 2;
        v2f b; b.x = W[(long)kb * dout + nc]; b.y = W[(long)(kb + 1) * dout + nc];
#pragma unroll
        for (int t = 0; t < MT; ++t) {
            const float* ap = A1 + (long)mrow[t] * lda1 + (kb - splitK);
            v2f a; a.x = ap[0]; a.y = ap[1];
            acc[t] = __builtin_amdgcn_wmma_f32_16x16x4_f32(false, a, false, b, (short)0,
                                                           acc[t], false, false);
        }
    }
#else
#pragma unroll
    for (int t = 0; t < MT; ++t)
        for (int r = 0; r < 8; ++r) {
            int row = tm * (16 * MT) + t * 16 + r + hi * 8;
            if (row >= nRows) row = nRows - 1;
            float s = 0.f;
            for (int k = 0; k < din; ++k) {
                float av = (k < splitK) ? A0[(long)row * lda0 + k]
                                        : A1[(long)row * lda1 + (k - splitK)];
                s += av * W[(long)k * dout + nc];
            }
            acc[t][r] = s;
        }
#endif

    float bv = bias ? bias[nc] : 0.f;
#pragma unroll
    for (int t = 0; t < MT; ++t) {
#pragma unroll
        for (int r = 0; r < 8; ++r) {
            int row = tm * (16 * MT) + t * 16 + r + hi * 8;  // D layout: VGPR r = M=r / M=r+8
            if (row >= nRows) continue;
            float v = acc[t][r] + bv;
            if (act == 1) v = v > 0.f ? v : 0.f;
            else if (act == 2) v = eluf(v);
            else if (act == 3) v = sigm(v);
            Out[(long)row * dout + nc] = v;
        }
    }
}

// ---------------------------------------------------------------- GAT helpers
__global__ void dots_kernel(const float* __restrict__ h, const float* __restrict__ as,
                            const float* __restrict__ ad, float* __restrict__ es,
                            float* __restrict__ ed, int N) {
    int i = blockIdx.x * blockDim.x + threadIdx.x;
    if (i >= N) return;
    float a = 0.f, b = 0.f;
#pragma unroll 8
    for (int c = 0; c < HID; ++c) { float v = h[(long)i * HID + c]; a += v * as[c]; b += v * ad[c]; }
    es[i] = a; ed[i] = b;
}

// idx < E: real edge (masked by dyn); idx >= E: self loop for node idx-E
__global__ void gat_max_kernel(const int* __restrict__ ei, const float* __restrict__ dyn,
                               const float* __restrict__ es, const float* __restrict__ ed,
                               unsigned* __restrict__ menc, int E, int N) {
    int idx = blockIdx.x * blockDim.x + threadIdx.x;
    if (idx >= E + N) return;
    int s, d;
    if (idx < E) { if (dyn[idx] < 0.2f) return; s = ei[idx]; d = ei[E + idx]; }
    else { s = d = idx - E; }
    atomicMax(&menc[d], enc_key(lrelu02(es[s] + ed[d])));
}

__global__ void gat_sum_kernel(const int* __restrict__ ei, const float* __restrict__ dyn,
                               const float* __restrict__ es, const float* __restrict__ ed,
                               const unsigned* __restrict__ menc, float* __restrict__ ssum,
                               int E, int N) {
    int idx = blockIdx.x * blockDim.x + threadIdx.x;
    if (idx >= E + N) return;
    int s, d;
    if (idx < E) { if (dyn[idx] < 0.2f) return; s = ei[idx]; d = ei[E + idx]; }
    else { s = d = idx - E; }
    float e = lrelu02(es[s] + ed[d]);
    atomicAdd(&ssum[d], expf(e - dec_key(menc[d])));
}

__global__ void gat_scatter_kernel(const int* __restrict__ ei, const float* __restrict__ dyn,
                                   const float* __restrict__ es, const float* __restrict__ ed,
                                   const unsigned* __restrict__ menc, const float* __restrict__ ssum,
                                   const float* __restrict__ h, float* __restrict__ agg,
                                   int E, int N) {
    int idx = blockIdx.x * blockDim.x + threadIdx.x;
    if (idx >= E + N) return;
    int s, d;
    if (idx < E) { if (dyn[idx] < 0.2f) return; s = ei[idx]; d = ei[E + idx]; }
    else { s = d = idx - E; }
    float e = lrelu02(es[s] + ed[d]);
    float alpha = expf(e - dec_key(menc[d])) / ssum[d];
    const float* hr = h + (long)s * HID;
    float* orow = agg + (long)d * HID;
#pragma unroll 8
    for (int c = 0; c < HID; ++c) atomicAdd(&orow[c], hr[c] * alpha);
}

// ---------------------------------------------------------------- BatchNorm
__global__ void bias_elu_bnreduce_kernel(float* __restrict__ x, const float* __restrict__ bias,
                                         float* __restrict__ red, long n64) {
    __shared__ float ls[HID], lq[HID];
    int t = threadIdx.x;
    if (t < HID) { ls[t] = 0.f; lq[t] = 0.f; }
    __syncthreads();
    long idx = (long)blockIdx.x * blockDim.x + t;
    if (idx < n64) {
        int c = (int)(idx & (HID - 1));
        float v = eluf(x[idx] + bias[c]);
        x[idx] = v;
        atomicAdd(&ls[c], v);
        atomicAdd(&lq[c], v * v);
    }
    __syncthreads();
    if (t < HID) { atomicAdd(&red[t], ls[t]); atomicAdd(&red[HID + t], lq[t]); }
}

__global__ void bn_apply_kernel(float* __restrict__ x, const float* __restrict__ red,
                                const float* __restrict__ g, const float* __restrict__ b,
                                float invN, long n64) {
    long idx = (long)blockIdx.x * blockDim.x + threadIdx.x;
    if (idx >= n64) return;
    int c = (int)(idx & (HID - 1));
    float mu = red[c] * invN;
    float var = red[HID + c] * invN - mu * mu;
    x[idx] = (x[idx] - mu) * rsqrtf(var + 1e-5f) * g[c] + b[c];
}

// ---------------------------------------------------------------- GCN
__global__ void gcn_deg_kernel(const int* __restrict__ ei, const float* __restrict__ dyn,
                               float* __restrict__ deg, int E, int N) {
    int idx = blockIdx.x * blockDim.x + threadIdx.x;
    if (idx >= E + N) return;
    int d; float w;
    if (idx < E) { w = dyn[idx]; if (w < 0.2f) return; d = ei[E + idx]; }
    else { w = 1.f; d = idx - E; }
    atomicAdd(&deg[d], w);
}

__global__ void gcn_scatter_kernel(const int* __restrict__ ei, const float* __restrict__ dyn,
                                   const float* __restrict__ deg, const float* __restrict__ h,
                                   float* __restrict__ agg, int E, int N) {
    int idx = blockIdx.x * blockDim.x + threadIdx.x;
    if (idx >= E + N) return;
    int s, d; float w;
    if (idx < E) { w = dyn[idx]; if (w < 0.2f) return; s = ei[idx]; d = ei[E + idx]; }
    else { w = 1.f; s = d = idx - E; }
    float ds_ = deg[s], dd_ = deg[d];
    float is_ = ds_ > 0.f ? rsqrtf(fmaxf(ds_, 1e-12f)) : 0.f;
    float id_ = dd_ > 0.f ? rsqrtf(fmaxf(dd_, 1e-12f)) : 0.f;
    float norm = is_ * w * id_;
    const float* hr = h + (long)s * HID;
    float* orow = agg + (long)d * HID;
#pragma unroll 8
    for (int c = 0; c < HID; ++c) atomicAdd(&orow[c], hr[c] * norm);
}

// x = LayerNorm(relu(x + b)) * g + beta, per row of 64
__global__ void gcn_post_kernel(float* __restrict__ x, const float* __restrict__ b,
                                const float* __restrict__ g, const float* __restrict__ beta,
                                int N) {
    int i = blockIdx.x * blockDim.x + threadIdx.x;
    if (i >= N) return;
    float v[HID]; float mu = 0.f;
#pragma unroll
    for (int c = 0; c < HID; ++c) { float t = x[(long)i * HID + c] + b[c]; t = t > 0.f ? t : 0.f; v[c] = t; mu += t; }
    mu *= (1.f / HID);
    float var = 0.f;
#pragma unroll
    for (int c = 0; c < HID; ++c) { float dd = v[c] - mu; var += dd * dd; }
    var *= (1.f / HID);
    float inv = rsqrtf(var + 1e-5f);
#pragma unroll
    for (int c = 0; c < HID; ++c) x[(long)i * HID + c] = (v[c] - mu) * inv * g[c] + beta[c];
}

// ---------------------------------------------------------------- gated output
__global__ void out_gate_kernel(const float* __restrict__ xl, const float* __restrict__ xg,
                                const float* __restrict__ gate, const float* __restrict__ ow,
                                const float* __restrict__ ob, float* __restrict__ out, int N) {
    int i = blockIdx.x * blockDim.x + threadIdx.x;
    if (i >= N) return;
    float acc = 0.f;
#pragma unroll 8
    for (int c = 0; c < HID; ++c) {
        float g = gate[(long)i * HID + c];
        acc += (g * xl[(long)i * HID + c] + (1.f - g) * xg[(long)i * HID + c]) * ow[c];
    }
    out[i] = acc + ob[0];
}

// ================================================================ host
static inline void launch_gemm(const float* A0, const float* A1, int splitK, int lda0, int lda1,
                               const float* W, const float* bias, float* Out,
                               int nRows, int din, int dout, int act, hipStream_t st) {
    int tiles = ((nRows + 16 * MT - 1) / (16 * MT)) * (dout / 16);
    int grid = (tiles + 7) / 8;   // 8 waves (wave32) per 256-thread block
    gemm_wmma_kernel<<<grid, 256, 0, st>>>(A0, A1, splitK, lda0, lda1, W, bias, Out,
                                           nRows, din, dout, act);
}
static inline void launch_fill(float* p, float v, long n, hipStream_t st) {
    fill_kernel<<<(int)((n + 255) / 256), 256, 0, st>>>(p, v, n);
}

extern "C" void kernel_launch(void* const* d_in, const int* in_sizes, int n_in,
                              void* d_out, int out_size, void* d_ws, size_t ws_size,
                              hipStream_t stream) {
    const int*   ei     = (const int*)  d_in[0];
    const float* noise  = (const float*)d_in[1];
    const float* coord  = (const float*)d_in[2];
    const float* x_loc  = (const float*)d_in[3];
    const float* x_glo  = (const float*)d_in[4];
    const float* ew_w1  = (const float*)d_in[5];
    const float* ew_b1  = (const float*)d_in[6];
    const float* ew_w2  = (const float*)d_in[7];
    const float* ew_b2  = (const float*)d_in[8];
    const float* gat_W[2]  = {(const float*)d_in[9],  (const float*)d_in[15]};
    const float* gat_as[2] = {(const float*)d_in[10], (const float*)d_in[16]};
    const float* gat_ad[2] = {(const float*)d_in[11], (const float*)d_in[17]};
    const float* gat_b[2]  = {(const float*)d_in[12], (const float*)d_in[18]};
    const float* bn_g[2]   = {(const float*)d_in[13], (const float*)d_in[19]};
    const float* bn_b[2]   = {(const float*)d_in[14], (const float*)d_in[20]};
    const float* gcn_W  = (const float*)d_in[21];
    const float* gcn_b  = (const float*)d_in[22];
    const float* ln_g   = (const float*)d_in[23];
    const float* ln_b   = (const float*)d_in[24];
    const float* la1_w  = (const float*)d_in[25];
    const float* la1_b  = (const float*)d_in[26];
    const float* la2_w  = (const float*)d_in[27];
    const float* la2_b  = (const float*)d_in[28];
    const float* ga1_w  = (const float*)d_in[29];
    const float* ga1_b  = (const float*)d_in[30];
    const float* ga2_w  = (const float*)d_in[31];
    const float* ga2_b  = (const float*)d_in[32];
    const float* fu1_w  = (const float*)d_in[33];
    const float* fu1_b  = (const float*)d_in[34];
    const float* fu2_w  = (const float*)d_in[35];
    const float* fu2_b  = (const float*)d_in[36];
    const float* gate_w = (const float*)d_in[37];
    const float* gate_b = (const float*)d_in[38];
    const float* out_w  = (const float*)d_in[39];
    const float* out_b  = (const float*)d_in[40];

    const int E = in_sizes[0] / 2;
    const int N = in_sizes[3] / 32;
    const long n64 = (long)N * HID;

    float* ws = (float*)d_ws;
    long o = 0;
    float* dyn   = ws + o; o += E;
    float* bufA  = ws + o; o += n64;   // h / hg / fusion-mid
    float* bufB  = ws + o; o += n64;   // gat0 agg -> xl0 -> XL (la2 out)
    float* bufC  = ws + o; o += n64;   // gat1 agg -> xl1 -> gcn agg -> xg -> gate
    float* bufD  = ws + o; o += n64;   // XG (ga2 out)
    float* buf32 = ws + o; o += (long)N * 32;
    float* es    = ws + o; o += N;
    float* ed    = ws + o; o += N;
    float* menc  = ws + o; o += N;     // reinterpreted as unsigned
    float* ssum  = ws + o; o += N;
    float* deg   = ws + o; o += N;
    float* red   = ws + o; o += 2 * HID;

    float* outv  = (float*)d_out;          // [N]
    float* fused = (float*)d_out + N;      // [N*64]

    const int gE  = (E + 255) / 256;
    const int gEN = (E + N + 255) / 256;
    const int gN  = (N + 255) / 256;
    const float invN = 1.f / (float)N;

    // edge weights
    edge_weight_kernel<<<gE, 256, 0, stream>>>(ei, noise, coord, ew_w1, ew_b1, ew_w2, ew_b2, dyn, E);

    // ---- local branch: 2x (GAT -> ELU -> BatchNorm)
    const float* xin = x_loc; int din = 32;
    float* aggbuf[2] = {bufB, bufC};
    for (int l = 0; l < 2; ++l) {
        float* agg = aggbuf[l];
        launch_gemm(xin, nullptr, din, din, 0, gat_W[l], nullptr, bufA, N, din, HID, 0, stream);
        dots_kernel<<<gN, 256, 0, stream>>>(bufA, gat_as[l], gat_ad[l], es, ed, N);
        launch_fill(menc, 0.0f, N, stream);       // enc(-inf) ~ 0
        launch_fill(ssum, 0.0f, N, stream);
        launch_fill(agg, 0.0f, n64, stream);
        gat_max_kernel<<<gEN, 256, 0, stream>>>(ei, dyn, es, ed, (unsigned*)menc, E, N);
        gat_sum_kernel<<<gEN, 256, 0, stream>>>(ei, dyn, es, ed, (const unsigned*)menc, ssum, E, N);
        gat_scatter_kernel<<<gEN, 256, 0, stream>>>(ei, dyn, es, ed, (const unsigned*)menc, ssum,
                                                    bufA, agg, E, N);
        launch_fill(red, 0.0f, 2 * HID, stream);
        bias_elu_bnreduce_kernel<<<(int)((n64 + 255) / 256), 256, 0, stream>>>(agg, gat_b[l], red, n64);
        bn_apply_kernel<<<(int)((n64 + 255) / 256), 256, 0, stream>>>(agg, red, bn_g[l], bn_b[l],
                                                                      invN, n64);
        xin = agg; din = HID;
    }
    // local adapter: XL = relu(xl1 @ la1 + b) @ la2 + b  -> bufB
    launch_gemm(bufC, nullptr, HID, HID, 0, la1_w, la1_b, buf32, N, HID, 32, 1, stream);
    launch_gemm(buf32, nullptr, 32, 32, 0, la2_w, la2_b, bufB, N, 32, HID, 0, stream);

    // ---- global branch: GCN -> ReLU -> LayerNorm
    launch_gemm(x_glo, nullptr, 32, 32, 0, gcn_W, nullptr, bufA, N, 32, HID, 0, stream);
    launch_fill(deg, 0.0f, N, stream);
    launch_fill(bufC, 0.0f, n64, stream);
    gcn_deg_kernel<<<gEN, 256, 0, stream>>>(ei, dyn, deg, E, N);
    gcn_scatter_kernel<<<gEN, 256, 0, stream>>>(ei, dyn, deg, bufA, bufC, E, N);
    gcn_post_kernel<<<gN, 256, 0, stream>>>(bufC, gcn_b, ln_g, ln_b, N);
    // global adapter: XG -> bufD
    launch_gemm(bufC, nullptr, HID, HID, 0, ga1_w, ga1_b, buf32, N, HID, 32, 1, stream);
    launch_gemm(buf32, nullptr, 32, 32, 0, ga2_w, ga2_b, bufD, N, 32, HID, 0, stream);

    // ---- fusion (virtual concat [XL|XG], din=128)
    launch_gemm(bufB, bufD, HID, HID, HID, fu1_w, fu1_b, bufA, N, 128, HID, 1, stream);
    launch_gemm(bufA, nullptr, HID, HID, 0, fu2_w, fu2_b, fused, N, HID, HID, 0, stream);
    // gate -> bufC
    launch_gemm(bufB, bufD, HID, HID, HID, gate_w, gate_b, bufC, N, 128, HID, 3, stream);
    // out = (gate*XL + (1-gate)*XG) @ out_w + out_b
    out_gate_kernel<<<gN, 256, 0, stream>>>(bufB, bufD, bufC, out_w, out_b, outv, N);

    (void)n_in; (void)out_size; (void)ws_size;
}